// Attention_27230092657507
// MI455X (gfx1250) — compile-verified
//
#include <hip/hip_runtime.h>
#include <hip/hip_bf16.h>
#include <math.h>

typedef __attribute__((ext_vector_type(2))) float v2f;
typedef __attribute__((ext_vector_type(8))) float v8f;

// Problem constants
// B=2, DIM=192, H=W=256, HEADS=4, WS=8 -> cp=48, d=64, Hn=Wn=32
// spatial S = 65536 per batch image

// ---------------------------------------------------------------------------
// GEMM for NCHW 1x1 conv:  out[b,o,s] = sum_c in[b,c,s] * w[o,c]
// M = output channels (A = weights), N = spatial (B = activations, coalesced)
// Wave tile: 16 (channels) x 64 (spatial), K step 4 via V_WMMA_F32_16X16X4_F32
// grid.x = 131072/512 = 256 spatial chunks (8 waves x 64), grid.y = Cout/16
// ---------------------------------------------------------------------------
__global__ __launch_bounds__(256) void gemm_nchw_wmma(
    const float* __restrict__ in, const float* __restrict__ wgt,
    float* __restrict__ out, int Cin, int Cout)
{
    const int lane = threadIdx.x & 31;
    const int wave = threadIdx.x >> 5;
    const int half = lane >> 4;        // 0: lanes 0-15, 1: lanes 16-31
    const int l16  = lane & 15;

    const int n0 = blockIdx.x * 512 + wave * 64;   // spatial base (never straddles batch)
    const int m0 = blockIdx.y * 16;                // output-channel base
    const int b  = n0 >> 16;
    const int s0 = n0 & 65535;

    const float* inb  = in  + (long)b * Cin * 65536 + s0;
    const float* wrow = wgt + (long)(m0 + l16) * Cin;   // A-frag row for this lane

    v8f acc[4];
    #pragma unroll
    for (int j = 0; j < 4; ++j)
        #pragma unroll
        for (int r = 0; r < 8; ++r) acc[j][r] = 0.0f;

    for (int kk = 0; kk < Cin; kk += 4) {
        const int k0 = kk + half * 2;
        // A fragment (weights): lane holds W[m0+l16][k0], W[m0+l16][k0+1]
        v2f a = *(const v2f*)(wrow + k0);
        const float* bk0 = inb + (long)k0 * 65536;
        const float* bk1 = bk0 + 65536;
        #pragma unroll
        for (int j = 0; j < 4; ++j) {
            // B fragment: lane holds In[k0][n], In[k0+1][n], n = s0+j*16+l16
            v2f bf;
            bf.x = bk0[j * 16 + l16];
            bf.y = bk1[j * 16 + l16];
            acc[j] = __builtin_amdgcn_wmma_f32_16x16x4_f32(
                false, a, false, bf, (short)0, acc[j], false, false);
        }
    }

    // D layout: lane holds (M = r + 8*half, N = l16) -> contiguous 16-float
    // spatial runs per half-wave per channel row: fully coalesced stores.
    #pragma unroll
    for (int j = 0; j < 4; ++j) {
        #pragma unroll
        for (int r = 0; r < 8; ++r) {
            const int o = m0 + r + 8 * half;
            out[(long)(b * Cout + o) * 65536 + s0 + j * 16 + l16] = acc[j][r];
        }
    }
}

// ---------------------------------------------------------------------------
// Depthwise 3x3 conv, SAME padding, groups = channels. Pure streaming stencil.
// ---------------------------------------------------------------------------
__global__ __launch_bounds__(256) void dwconv3x3(
    const float* __restrict__ in, const float* __restrict__ w,
    float* __restrict__ out)
{
    const long idx = (long)blockIdx.x * 256 + threadIdx.x;  // < 2*576*65536
    const int  xc  = (int)(idx & 255);
    const int  yc  = (int)((idx >> 8) & 255);
    const long plane = idx >> 16;               // b*576 + c  (uniform per block)
    const int  c   = (int)(plane % 576);

    const float* wp = w + c * 9;
    const float* ip = in + (plane << 16);

    float sum = 0.0f;
    #pragma unroll
    for (int dy = -1; dy <= 1; ++dy) {
        const int y = yc + dy;
        if (y < 0 || y > 255) continue;
        #pragma unroll
        for (int dx = -1; dx <= 1; ++dx) {
            const int x = xc + dx;
            if (x < 0 || x > 255) continue;
            sum = fmaf(ip[y * 256 + x], wp[(dy + 1) * 3 + (dx + 1)], sum);
        }
    }
    out[idx] = sum;
}

// ---------------------------------------------------------------------------
// Windowed channel attention: one wave per (window, head). 8192 blocks.
//   q,k,v : [48 x 64] fp32 in LDS (stride 68 -> conflict-free column reads)
//   attn  : [48 x 48] in LDS (stride 52)
//   qn@kn^T and attn@v via V_WMMA_F32_16X16X4_F32
// ---------------------------------------------------------------------------
__global__ __launch_bounds__(32) void win_attn(
    const float* __restrict__ qkv, const float* __restrict__ temp,
    float* __restrict__ out)
{
    __shared__ float sQ[48][68];
    __shared__ float sK[48][68];
    __shared__ float sV[48][68];
    __shared__ float sA[48][52];

    const int lane = threadIdx.x;
    const int half = lane >> 4;
    const int l16  = lane & 15;

    const int win = blockIdx.x >> 2;   // 0..2047
    const int h   = blockIdx.x & 3;    // head
    const int b   = win >> 10;
    const int rem = win & 1023;
    const int wy  = rem >> 5;          // window row (Hn=32)
    const int wx  = rem & 31;          // window col (Wn=32)

    // ---- load q,k,v window tiles (float4 vectorized) ----
    for (int t = lane; t < 768; t += 32) {          // 48 rows x 16 float4
        const int cp = t >> 4;
        const int u  = t & 15;
        const int p1 = u >> 1;
        const int p2 = (u & 1) * 4;
        const int lo = p1 * 8 + p2;
        #pragma unroll
        for (int p = 0; p < 3; ++p) {
            const long g = (((long)(b * 576 + p * 192 + h * 48 + cp) * 256)
                            + wy * 8 + p1) * 256 + wx * 8 + p2;
            const float4 val = *(const float4*)(qkv + g);
            float* dst = (p == 0) ? &sQ[cp][lo] : (p == 1) ? &sK[cp][lo] : &sV[cp][lo];
            dst[0] = val.x; dst[1] = val.y; dst[2] = val.z; dst[3] = val.w;
        }
    }
    __syncthreads();

    // ---- l2-normalize rows of q and k over d=64 (eps = 1e-12) ----
    for (int r = lane; r < 96; r += 32) {
        float* row = (r < 48) ? sQ[r] : sK[r - 48];
        float ss = 0.0f;
        for (int d = 0; d < 64; ++d) ss = fmaf(row[d], row[d], ss);
        const float inv = 1.0f / fmaxf(sqrtf(ss), 1e-12f);
        for (int d = 0; d < 64; ++d) row[d] *= inv;
    }
    __syncthreads();

    // ---- attn = qn @ kn^T  (48x48, K=64) ----
    v8f accA[3][3];
    #pragma unroll
    for (int i = 0; i < 3; ++i)
        #pragma unroll
        for (int j = 0; j < 3; ++j)
            #pragma unroll
            for (int r = 0; r < 8; ++r) accA[i][j][r] = 0.0f;

    for (int kk = 0; kk < 64; kk += 4) {
        const int k0 = kk + half * 2;
        v2f af[3], bf[3];
        #pragma unroll
        for (int i = 0; i < 3; ++i) {
            const int m = i * 16 + l16;
            af[i].x = sQ[m][k0]; af[i].y = sQ[m][k0 + 1];
        }
        #pragma unroll
        for (int j = 0; j < 3; ++j) {
            const int n = j * 16 + l16;                  // B[k][n] = kn[n][k]
            bf[j].x = sK[n][k0]; bf[j].y = sK[n][k0 + 1];
        }
        #pragma unroll
        for (int i = 0; i < 3; ++i)
            #pragma unroll
            for (int j = 0; j < 3; ++j)
                accA[i][j] = __builtin_amdgcn_wmma_f32_16x16x4_f32(
                    false, af[i], false, bf[j], (short)0, accA[i][j], false, false);
    }

    // temperature scale + spill attn to LDS for softmax
    const float tv = temp[h];
    #pragma unroll
    for (int i = 0; i < 3; ++i)
        #pragma unroll
        for (int j = 0; j < 3; ++j)
            #pragma unroll
            for (int r = 0; r < 8; ++r)
                sA[i * 16 + r + 8 * half][j * 16 + l16] = accA[i][j][r] * tv;
    __syncthreads();

    // ---- row softmax over 48 ----
    for (int r = lane; r < 48; r += 32) {
        float mx = -3.402823466e+38f;
        for (int e = 0; e < 48; ++e) mx = fmaxf(mx, sA[r][e]);
        float sum = 0.0f;
        for (int e = 0; e < 48; ++e) {
            const float ex = __expf(sA[r][e] - mx);
            sA[r][e] = ex;
            sum += ex;
        }
        const float inv = 1.0f / sum;
        for (int e = 0; e < 48; ++e) sA[r][e] *= inv;
    }
    __syncthreads();

    // ---- out = attn @ v  (48x64, K=48) ----
    v8f accO[3][4];
    #pragma unroll
    for (int i = 0; i < 3; ++i)
        #pragma unroll
        for (int j = 0; j < 4; ++j)
            #pragma unroll
            for (int r = 0; r < 8; ++r) accO[i][j][r] = 0.0f;

    for (int kk = 0; kk < 48; kk += 4) {
        const int k0 = kk + half * 2;
        v2f af[3], bf[4];
        #pragma unroll
        for (int i = 0; i < 3; ++i) {
            const int m = i * 16 + l16;
            af[i].x = sA[m][k0]; af[i].y = sA[m][k0 + 1];
        }
        #pragma unroll
        for (int j = 0; j < 4; ++j) {
            const int n = j * 16 + l16;                  // B[k][n] = v[k][n]
            bf[j].x = sV[k0][n]; bf[j].y = sV[k0 + 1][n];
        }
        #pragma unroll
        for (int i = 0; i < 3; ++i)
            #pragma unroll
            for (int j = 0; j < 4; ++j)
                accO[i][j] = __builtin_amdgcn_wmma_f32_16x16x4_f32(
                    false, af[i], false, bf[j], (short)0, accO[i][j], false, false);
    }

    // ---- scatter back to NCHW windowed layout ----
    #pragma unroll
    for (int i = 0; i < 3; ++i)
        #pragma unroll
        for (int j = 0; j < 4; ++j)
            #pragma unroll
            for (int r = 0; r < 8; ++r) {
                const int cp = i * 16 + r + 8 * half;
                const int d  = j * 16 + l16;
                const int p1 = d >> 3, p2 = d & 7;
                const long g = (((long)(b * 192 + h * 48 + cp) * 256)
                                + wy * 8 + p1) * 256 + wx * 8 + p2;
                out[g] = accO[i][j][r];
            }
}

// ---------------------------------------------------------------------------
extern "C" void kernel_launch(void* const* d_in, const int* in_sizes, int n_in,
                              void* d_out, int out_size, void* d_ws, size_t ws_size,
                              hipStream_t stream)
{
    const float* x      = (const float*)d_in[0];  // [2,192,256,256]
    const float* qkv_w  = (const float*)d_in[1];  // [576,192]
    const float* dw_w   = (const float*)d_in[2];  // [576,1,3,3]
    const float* proj_w = (const float*)d_in[3];  // [192,192]
    const float* temp   = (const float*)d_in[4];  // [4,1,1]
    float* outp = (float*)d_out;                  // [2,192,256,256]

    float* buf0 = (float*)d_ws;                          // qkv, later attn out
    float* buf1 = buf0 + (size_t)2 * 576 * 65536;        // qkv after dw conv

    // 1) qkv 1x1 conv: GEMM [131072 x 576 x 192]
    gemm_nchw_wmma<<<dim3(256, 36), 256, 0, stream>>>(x, qkv_w, buf0, 192, 576);
    // 2) depthwise 3x3 conv (SAME)
    dwconv3x3<<<dim3(294912), 256, 0, stream>>>(buf0, dw_w, buf1);
    // 3) windowed channel attention; reuses buf0 (qkv dead) for output
    win_attn<<<dim3(8192), 32, 0, stream>>>(buf1, temp, buf0);
    // 4) proj 1x1 conv: GEMM [131072 x 192 x 192]
    gemm_nchw_wmma<<<dim3(256, 12), 256, 0, stream>>>(buf0, proj_w, outp, 192, 192);
}